// self_LSTM_sback_90907277787372
// MI455X (gfx1250) — compile-verified
//
#include <hip/hip_runtime.h>
#include <math.h>

// ---------------------------------------------------------------------------
// Persistent single-WGP LSTM+online-softmax-attention kernel for MI455X.
//   gates = [h | x_t | 1 | 0] @ Waug^T   via V_WMMA_F32_16X16X4_F32 (fp32)
//   attention kept as running (m, D, N[h]) per batch (online softmax) -> no
//   history buffer, O(T*H) instead of O(T^2*H).
// All weights + state resident in LDS (~295 KB < 320 KB WGP LDS).
// This revision batches the four B-fragment DS loads per K-chunk into
// distinct registers so the DS->WMMA pipeline overlaps instead of
// serializing on s_wait_dscnt 0 before every WMMA.
// ---------------------------------------------------------------------------

typedef float v2f __attribute__((ext_vector_type(2)));
typedef float v8f __attribute__((ext_vector_type(8)));

#define BB   16
#define TT   256
#define HH   128
#define CC   10
#define KAUG 132            // 128 h + x + 1(bias) + 2 pad  (multiple of 4)
#define NKC  (KAUG / 4)     // 33 K-chunks of 4
#define HSTR 134            // h_lds row stride (floats), keeps 8B alignment
#define NTILES 32           // 512 gate columns / 16

// LDS partition (in floats)
#define OFF_BSWZ 0
#define SZ_BSWZ  (NTILES * NKC * 64)        // 67584  (264 KB) swizzled Waug
#define OFF_H    (OFF_BSWZ + SZ_BSWZ)
#define SZ_H     (BB * HSTR)                // 2144   h_aug rows
#define OFF_Y    (OFF_H + SZ_H)
#define SZ_Y     (BB * HH)                  // 2048   y = h_new + 2*attn
#define OFF_N    (OFF_Y + SZ_Y)
#define SZ_N     (BB * HH)                  // 2048   online-softmax numerator
#define OFF_FCW  (OFF_N + SZ_N)
#define SZ_FCW   (CC * HH)                  // 1280
#define OFF_WTO  (OFF_FCW + SZ_FCW)
#define SZ_WTO   (HH)                       // 128    wt_o
#define OFF_SP   (OFF_WTO + SZ_WTO)
#define SZ_SP    (8 * BB)                   // 128    per-wave score partials
#define OFF_M    (OFF_SP + SZ_SP)           // 16     running max
#define OFF_D    (OFF_M + BB)               // 16     running denom
#define OFF_ID   (OFF_D + BB)               // 16     1/denom
#define LDS_FLOATS (OFF_ID + BB)
#define LDS_BYTES  ((size_t)LDS_FLOATS * sizeof(float))

__global__ __launch_bounds__(256, 1)
void lstm_attn_persist(const float* __restrict__ x,      // (B,T)
                       const float* __restrict__ w_ih,   // (4H,1)
                       const float* __restrict__ w_hh,   // (4H,H)
                       const float* __restrict__ b_ih,   // (4H)
                       const float* __restrict__ b_hh,   // (4H)
                       const float* __restrict__ w_t,    // (2H,1)
                       const float* __restrict__ fc_w,   // (C,H)
                       const float* __restrict__ fc_b,   // (C)
                       float* __restrict__ out)          // (B,T,C)
{
  extern __shared__ float sm[];
  const int tid  = threadIdx.x;
  const int lane = tid & 31;
  const int w    = tid >> 5;      // wave id 0..7 -> hidden cols [16w,16w+16)
  const int row  = lane & 15;
  const int hi   = lane >> 4;

  // ------------------------- init: build LDS-resident state ----------------
  // Swizzled augmented-B fragments: frag(nt,kc) = 64 consecutive floats,
  // lane L supplies (B[k0+2*(L/16)][n0+L%16], B[k0+1+2*(L/16)][n0+L%16]).
  for (int idx = tid; idx < SZ_BSWZ; idx += 256) {
    int f = idx >> 6, u = idx & 63;
    int l = u >> 1, v = u & 1;
    int nt = f / NKC, kc = f - nt * NKC;
    int k = kc * 4 + ((l >> 4) << 1) + v;
    int n = nt * 16 + (l & 15);          // gate column j = g*128 + hcol
    float val = 0.f;
    if (k < HH)           val = w_hh[n * HH + k];
    else if (k == HH)     val = w_ih[n];                 // scalar-x row
    else if (k == HH + 1) val = b_ih[n] + b_hh[n];       // bias row
    sm[OFF_BSWZ + idx] = val;
  }
  for (int idx = tid; idx < SZ_H; idx += 256) {
    int b = idx / HSTR, col = idx - b * HSTR;
    float val = 0.f;
    if (col == HH)          val = x[b * TT + 0];   // x_0 in aug column
    else if (col == HH + 1) val = 1.f;             // bias lane
    sm[OFF_H + idx] = val;
  }
  for (int idx = tid; idx < SZ_N; idx += 256)   sm[OFF_N + idx]   = 0.f;
  for (int idx = tid; idx < SZ_FCW; idx += 256) sm[OFF_FCW + idx] = fc_w[idx];
  for (int idx = tid; idx < HH; idx += 256)     sm[OFF_WTO + idx] = w_t[HH + idx];
  if (tid < BB) {  // slot 0 (zeros, score 0) pre-folded: m=0, D=1, N=0
    sm[OFF_M + tid] = 0.f; sm[OFF_D + tid] = 1.f; sm[OFF_ID + tid] = 1.f;
  }
  __syncthreads();

  float c_reg[8];                        // cell state, resident in VGPRs
  #pragma unroll
  for (int r = 0; r < 8; ++r) c_reg[r] = 0.f;

  const int   h_idx = (w << 4) + row;    // this lane's hidden column
  const float wto   = sm[OFF_WTO + h_idx];
  const v8f   vzero = {0.f, 0.f, 0.f, 0.f, 0.f, 0.f, 0.f, 0.f};

  for (int i = 0; i < TT; ++i) {
    // ---- P1: gates via fp32 WMMA, LSTM cell, attention read, score ----
    v8f acc0 = vzero, acc1 = vzero, acc2 = vzero, acc3 = vzero;

    const int koff  = hi << 1;
    const int arow  = OFF_H + row * HSTR + koff;
    const int bbase = OFF_BSWZ + (w * NKC << 6) + (lane << 1);

    for (int kc = 0; kc < NKC; ++kc) {
      // Issue all five DS loads into distinct registers, then the 4 WMMAs,
      // so waits can retire progressively instead of dscnt==0 per WMMA.
      const float* ap  = &sm[arow + kc * 4];
      const float* bp0 = &sm[bbase + (kc << 6)];
      const float* bp1 = bp0 + (8 * NKC << 6);
      const float* bp2 = bp0 + (16 * NKC << 6);
      const float* bp3 = bp0 + (24 * NKC << 6);
      v2f a;  a.x  = ap[0];  a.y  = ap[1];
      v2f b0; b0.x = bp0[0]; b0.y = bp0[1];
      v2f b1; b1.x = bp1[0]; b1.y = bp1[1];
      v2f b2; b2.x = bp2[0]; b2.y = bp2[1];
      v2f b3; b3.x = bp3[0]; b3.y = bp3[1];
      acc0 = __builtin_amdgcn_wmma_f32_16x16x4_f32(false, a, false, b0,
                                                   (short)0, acc0, false, false);
      acc1 = __builtin_amdgcn_wmma_f32_16x16x4_f32(false, a, false, b1,
                                                   (short)0, acc1, false, false);
      acc2 = __builtin_amdgcn_wmma_f32_16x16x4_f32(false, a, false, b2,
                                                   (short)0, acc2, false, false);
      acc3 = __builtin_amdgcn_wmma_f32_16x16x4_f32(false, a, false, b3,
                                                   (short)0, acc3, false, false);
    }

    float spart[8];
    #pragma unroll
    for (int r = 0; r < 8; ++r) {
      float ig = acc0[r], fg = acc1[r], gg = acc2[r], og = acc3[r];
      float si = 1.f / (1.f + expf(-ig));
      float sf = 1.f / (1.f + expf(-fg));
      float so = 1.f / (1.f + expf(-og));
      float cn = sf * c_reg[r] + si * tanhf(gg);
      c_reg[r] = cn;
      float hn = so * tanhf(cn);
      int   b  = r + (hi << 3);
      float attn = sm[OFF_N + b * HH + h_idx] * sm[OFF_ID + b];
      float hout = hn + attn;               // recurrent/stored state
      sm[OFF_H + b * HSTR + h_idx] = hout;  // feeds next step's A operand
      sm[OFF_Y + b * HH + h_idx]   = hn + 2.f * attn;  // scan output
      spart[r] = tanhf(hout) * wto;         // slot-score contribution
    }
    // deterministic cross-lane reduce over the 16 lanes sharing a batch set
    #pragma unroll
    for (int r = 0; r < 8; ++r) {
      float v = spart[r];
      v += __shfl_xor(v, 1, 16);
      v += __shfl_xor(v, 2, 16);
      v += __shfl_xor(v, 4, 16);
      v += __shfl_xor(v, 8, 16);
      if (row == 0) sm[OFF_SP + w * 16 + (hi << 3) + r] = v;
    }
    __syncthreads();

    // ---- P2: FC output + online-softmax numerator update ----
    if (tid < BB * CC) {
      int b = tid / CC, cl = tid - b * CC;
      float accv = fc_b[cl];
      const float* yr = &sm[OFF_Y + b * HH];
      const float* fr = &sm[OFF_FCW + cl * HH];
      for (int h = 0; h < HH; ++h) accv = fmaf(yr[h], fr[h], accv);
      out[(b * TT + i) * CC + cl] = accv;
    }
    {
      int b = tid >> 4, h0 = (tid & 15) << 3;
      float s = 0.f;
      #pragma unroll
      for (int ww = 0; ww < 8; ++ww) s += sm[OFF_SP + ww * 16 + b];
      float m_old = sm[OFF_M + b];
      float mn    = fmaxf(m_old, s);
      float scale = expf(m_old - mn);
      float e     = expf(s - mn);
      #pragma unroll
      for (int j = 0; j < 8; ++j) {
        int   h = h0 + j;
        float n = sm[OFF_N + b * HH + h];
        sm[OFF_N + b * HH + h] = n * scale + e * sm[OFF_H + b * HSTR + h];
      }
    }
    __syncthreads();

    // ---- P3: denominator/max update + stage x_{i+1} ----
    if (tid < BB) {
      int b = tid;
      float s = 0.f;
      #pragma unroll
      for (int ww = 0; ww < 8; ++ww) s += sm[OFF_SP + ww * 16 + b];
      float m_old = sm[OFF_M + b];
      float mn    = fmaxf(m_old, s);
      float scale = expf(m_old - mn);
      float e     = expf(s - mn);
      float Dn    = sm[OFF_D + b] * scale + e;
      sm[OFF_D + b]  = Dn;
      sm[OFF_ID + b] = 1.f / Dn;
      sm[OFF_M + b]  = mn;
    } else if (tid < 2 * BB) {
      int b = tid - BB;
      sm[OFF_H + b * HSTR + HH] = (i + 1 < TT) ? x[b * TT + i + 1] : 0.f;
    }
    __syncthreads();
  }
}

extern "C" void kernel_launch(void* const* d_in, const int* in_sizes, int n_in,
                              void* d_out, int out_size, void* d_ws, size_t ws_size,
                              hipStream_t stream) {
  (void)in_sizes; (void)n_in; (void)out_size; (void)d_ws; (void)ws_size;
  const float* x    = (const float*)d_in[0];
  const float* wih  = (const float*)d_in[1];
  const float* whh  = (const float*)d_in[2];
  const float* bih  = (const float*)d_in[3];
  const float* bhh  = (const float*)d_in[4];
  const float* wt   = (const float*)d_in[5];
  const float* fcw  = (const float*)d_in[6];
  const float* fcb  = (const float*)d_in[7];
  // d_in[8] = truncate_length (forward no-op), d_in[9] = attn_every_k (== 1)
  float* out = (float*)d_out;

  hipLaunchKernelGGL(lstm_attn_persist, dim3(1), dim3(256), LDS_BYTES, stream,
                     x, wih, whh, bih, bhh, wt, fcw, fcb, out);
}